// DPhysics_6614249636453
// MI455X (gfx1250) — compile-verified
//
#include <hip/hip_runtime.h>
#include <cstdint>
#include <cstddef>

namespace {
constexpr int   kB   = 64;
constexpr int   kP   = 256;
constexpr int   kW   = 128;
constexpr int   kHW  = 128 * 128;
constexpr int   kT   = 500;
constexpr float kDT  = 0.01f;
constexpr float kMG  = 40.0f * 9.81f;   // 392.4
constexpr float kEPS = 1e-6f;

// Output sections (flat float offsets), in reference return order:
// Xs(B,T,3) Xds(B,T,3) Rs(B,T,3,3) Oms(B,T,3) Fs(B,T,P,3) Ff(B,T,P,3)
constexpr size_t oXs = 0;
constexpr size_t oXd = (size_t)kB * kT * 3;              //  96000
constexpr size_t oR  = oXd + (size_t)kB * kT * 3;        // 192000
constexpr size_t oOm = oR  + (size_t)kB * kT * 9;        // 480000
constexpr size_t oFs = oOm + (size_t)kB * kT * 3;        // 576000
constexpr size_t oFf = oFs + (size_t)kB * kT * kP * 3;   // 25152000
}

__device__ __forceinline__ float clipf(float v, float lo, float hi) {
    return fminf(fmaxf(v, lo), hi);
}

// deterministic wave32 sum
__device__ __forceinline__ float wsum32(float v) {
    v += __shfl_xor(v, 16, 32);
    v += __shfl_xor(v,  8, 32);
    v += __shfl_xor(v,  4, 32);
    v += __shfl_xor(v,  2, 32);
    v += __shfl_xor(v,  1, 32);
    return v;
}

// CDNA5 async global->LDS copy, 16B per lane (tracked by ASYNCcnt).
// ldsAddr is the raw 32-bit LDS byte address carried in the VDST vgpr.
__device__ __forceinline__ void async_g2l_b128(uint32_t ldsAddr, const void* gaddr) {
    asm volatile("global_load_async_to_lds_b128 %0, %1, off"
                 :: "v"(ldsAddr), "v"(gaddr) : "memory");
}

__device__ __forceinline__ void wait_asynccnt0() {
    asm volatile("s_wait_asynccnt 0x0" ::: "memory");
}

__global__ __launch_bounds__(256)
void dphysics_sim(const float* __restrict__ zg,   // (B,128,128)
                  const float* __restrict__ kg,   // stiffness
                  const float* __restrict__ cg,   // damping
                  const float* __restrict__ mug,  // friction
                  const float* __restrict__ ctrl, // (B,T,2)
                  const float* __restrict__ xpts, // (P,3)
                  float* __restrict__ out) {
    extern __shared__ float smem[];
    float* s_z   = smem;
    float* s_k   = smem + 1 * kHW;
    float* s_c   = smem + 2 * kHW;
    float* s_mu  = smem + 3 * kHW;
    float* s_red = smem + 4 * kHW;   // 8 waves * 6 partials

    const int b    = blockIdx.x;
    const int tid  = threadIdx.x;    // point index p
    const int lane = tid & 31;
    const int wave = tid >> 5;

    // ---- stage all 4 terrain grids for this batch into LDS (async path) ----
    {
        // single generic->LDS address conversion, integer math afterwards
        const uint32_t lbase = (uint32_t)(uintptr_t)smem;
        const float4*  g0 = (const float4*)(zg  + (size_t)b * kHW);
        const float4*  g1 = (const float4*)(kg  + (size_t)b * kHW);
        const float4*  g2 = (const float4*)(cg  + (size_t)b * kHW);
        const float4*  g3 = (const float4*)(mug + (size_t)b * kHW);
        #pragma unroll 1
        for (int i = tid; i < kHW / 4; i += 256) {
            const uint32_t l = lbase + (uint32_t)i * 16u;
            async_g2l_b128(l + 0u * (kHW * 4u), g0 + i);
            async_g2l_b128(l + 1u * (kHW * 4u), g1 + i);
            async_g2l_b128(l + 2u * (kHW * 4u), g2 + i);
            async_g2l_b128(l + 3u * (kHW * 4u), g3 + i);
        }
        wait_asynccnt0();
    }
    __syncthreads();

    // per-thread constant: body-frame point
    const float px = xpts[3 * tid + 0];
    const float py = xpts[3 * tid + 1];
    const float pz = xpts[3 * tid + 2];
    const float sideSgn = (py < 0.0f) ? -1.0f : 1.0f;  // track_vels: v + sgn*w*LY/2

    // broadcast state, kept identically in every thread's registers
    float x0 = 0.0f, x1 = 0.0f, x2 = 0.2f;
    float xd0 = 0.0f, xd1 = 0.0f, xd2 = 0.0f;
    float om0 = 0.0f, om1 = 0.0f, om2 = 0.0f;
    float R[3][3] = {{1.f,0.f,0.f},{0.f,1.f,0.f},{0.f,0.f,1.f}};

    #pragma unroll 1
    for (int t = 0; t < kT; ++t) {
        const size_t bt = (size_t)b * kT + t;

        // rel = R * p ; xp = rel + x ; xdp = xd + omega x rel
        const float r0 = R[0][0]*px + R[0][1]*py + R[0][2]*pz;
        const float r1 = R[1][0]*px + R[1][1]*py + R[1][2]*pz;
        const float r2 = R[2][0]*px + R[2][1]*py + R[2][2]*pz;
        const float xpx = r0 + x0, xpy = r1 + x1, xpz = r2 + x2;
        const float vx = xd0 + (om1 * r2 - om2 * r1);
        const float vy = xd1 + (om2 * r0 - om0 * r2);
        const float vz = xd2 + (om0 * r1 - om1 * r0);

        // bilinear interpolation setup (shared across 4 grids)
        const float gx = (xpx + 6.4f) * 10.0f;
        const float gy = (xpy + 6.4f) * 10.0f;
        int xi = (int)gx; xi = xi < 0 ? 0 : (xi > kW - 2 ? kW - 2 : xi);
        int yi = (int)gy; yi = yi < 0 ? 0 : (yi > kW - 2 ? kW - 2 : yi);
        const float xf = gx - (float)xi, yf = gy - (float)yi;
        const int i00 = yi * kW + xi;
        const float w00 = (1.f - xf) * (1.f - yf);
        const float w01 = (1.f - xf) * yf;
        const float w10 = xf * (1.f - yf);
        const float w11 = xf * yf;

        const float z00 = s_z[i00],          z01 = s_z[i00 + kW];
        const float z10 = s_z[i00 + 1],      z11 = s_z[i00 + kW + 1];
        const float zq  = z00*w00 + z01*w01 + z10*w10 + z11*w11;
        const float kq  = s_k [i00]*w00 + s_k [i00+kW]*w01 + s_k [i00+1]*w10 + s_k [i00+kW+1]*w11;
        const float cq  = s_c [i00]*w00 + s_c [i00+kW]*w01 + s_c [i00+1]*w10 + s_c [i00+kW+1]*w11;
        const float muq = s_mu[i00]*w00 + s_mu[i00+kW]*w01 + s_mu[i00+1]*w10 + s_mu[i00+kW+1]*w11;

        // surface normal
        const float dzdx = ((z10 - z00) * (1.f - yf) + (z11 - z01) * yf) * 10.0f;
        const float dzdy = ((z01 - z00) * (1.f - xf) + (z11 - z10) * xf) * 10.0f;
        const float nn   = fmaxf(sqrtf(dzdx*dzdx + dzdy*dzdy + 1.0f), kEPS);
        const float nx = -dzdx / nn, ny = -dzdy / nn, nz = 1.0f / nn;

        // spring force (clipped)
        const float dh  = xpz - zq;
        const float cont = 1.0f / (1.0f + expf(10.0f * dh));   // sigmoid(-10*dh)
        const float xdn = vx*nx + vy*ny + vz*nz;
        const float coef = -(kq * dh + cq * xdn) * cont * (1.0f / (float)kP);
        const float Fsx = clipf(coef * nx, -kMG, kMG);
        const float Fsy = clipf(coef * ny, -kMG, kMG);
        const float Fsz = clipf(coef * nz, -kMG, kMG);
        const float Nmag = sqrtf(Fsx*Fsx + Fsy*Fsy + Fsz*Fsz);

        // thrust = normalize(R[:,0])
        const float tn = fmaxf(sqrtf(R[0][0]*R[0][0] + R[1][0]*R[1][0] + R[2][0]*R[2][0]), kEPS);
        const float tx = R[0][0]/tn, ty = R[1][0]/tn, tz = R[2][0]/tn;

        // command velocity for this point's track side
        const float cv = ctrl[bt * 2 + 0];
        const float cw = ctrl[bt * 2 + 1];
        const float trackv = cv + sideSgn * cw * 0.25f;   // LY/2 = 0.25
        const float mcx = clipf(muq * trackv * tx, -2.0f, 2.0f);
        const float mcy = clipf(muq * trackv * ty, -2.0f, 2.0f);
        const float mcz = clipf(muq * trackv * tz, -2.0f, 2.0f);

        // friction: tangential velocity error
        const float dvx = mcx - vx, dvy = mcy - vy, dvz = mcz - vz;
        const float dvn = dvx*nx + dvy*ny + dvz*nz;
        const float Ffx = clipf(Nmag * (dvx - dvn*nx), -kMG, kMG);
        const float Ffy = clipf(Nmag * (dvy - dvn*ny), -kMG, kMG);
        const float Ffz = clipf(Nmag * (dvz - dvn*nz), -kMG, kMG);

        // per-point outputs: 196 MB write-once stream -> non-temporal stores
        {
            float* pFs = out + oFs + (bt * kP + tid) * 3;
            float* pFf = out + oFf + (bt * kP + tid) * 3;
            __builtin_nontemporal_store(Fsx, pFs + 0);
            __builtin_nontemporal_store(Fsy, pFs + 1);
            __builtin_nontemporal_store(Fsz, pFs + 2);
            __builtin_nontemporal_store(Ffx, pFf + 0);
            __builtin_nontemporal_store(Ffy, pFf + 1);
            __builtin_nontemporal_store(Ffz, pFf + 2);
        }

        // reduce: total force (3) + torque (3) over 256 points
        const float Ftx = Fsx + Ffx, Fty = Fsy + Ffy, Ftz = Fsz + Ffz;
        float red[6];
        red[0] = Ftx; red[1] = Fty; red[2] = Ftz;
        red[3] = r1 * Ftz - r2 * Fty;
        red[4] = r2 * Ftx - r0 * Ftz;
        red[5] = r0 * Fty - r1 * Ftx;
        #pragma unroll
        for (int q = 0; q < 6; ++q) {
            const float s = wsum32(red[q]);
            if (lane == 0) s_red[wave * 6 + q] = s;
        }
        __syncthreads();
        float tot[6];
        #pragma unroll
        for (int q = 0; q < 6; ++q) {
            float s = 0.0f;
            #pragma unroll
            for (int wv = 0; wv < 8; ++wv) s += s_red[wv * 6 + q];
            tot[q] = s;
        }
        __syncthreads();   // s_red reusable next step

        // ---- uniform state update (every thread computes identically) ----
        const float od0 = clipf(tot[3] * 0.2f, -2.0f, 2.0f);
        const float od1 = clipf(tot[4] * 0.2f, -2.0f, 2.0f);
        const float od2 = clipf(tot[5] * 0.2f, -2.0f, 2.0f);
        xd0 += (tot[0]            ) * (kDT / 40.0f);
        xd1 += (tot[1]            ) * (kDT / 40.0f);
        xd2 += (tot[2] - kMG      ) * (kDT / 40.0f);
        x0  += xd0 * kDT;  x1 += xd1 * kDT;  x2 += xd2 * kDT;
        om0 += od0 * kDT;  om1 += od1 * kDT; om2 += od2 * kDT;

        // R = R @ (I + sa*K + sb*(w w^T - |w|^2 I)),  K = skew(omega)
        const float oo    = om0*om0 + om1*om1 + om2*om2;
        const float theta = sqrtf(oo);
        const float th    = fmaxf(theta, kEPS);
        const float sa    = sinf(theta * kDT) / th;
        const float sb    = (1.0f - cosf(theta * kDT)) / (th * th);
        float Rot[3][3];
        Rot[0][0] = 1.0f + sb*(om0*om0 - oo);
        Rot[0][1] = -sa*om2 + sb*(om0*om1);
        Rot[0][2] =  sa*om1 + sb*(om0*om2);
        Rot[1][0] =  sa*om2 + sb*(om1*om0);
        Rot[1][1] = 1.0f + sb*(om1*om1 - oo);
        Rot[1][2] = -sa*om0 + sb*(om1*om2);
        Rot[2][0] = -sa*om1 + sb*(om2*om0);
        Rot[2][1] =  sa*om0 + sb*(om2*om1);
        Rot[2][2] = 1.0f + sb*(om2*om2 - oo);
        float Rn[3][3];
        #pragma unroll
        for (int i = 0; i < 3; ++i)
            #pragma unroll
            for (int j = 0; j < 3; ++j)
                Rn[i][j] = R[i][0]*Rot[0][j] + R[i][1]*Rot[1][j] + R[i][2]*Rot[2][j];
        #pragma unroll
        for (int i = 0; i < 3; ++i)
            #pragma unroll
            for (int j = 0; j < 3; ++j)
                R[i][j] = Rn[i][j];

        // state outputs (post-update values, matching the scan's ys)
        if (tid == 0) {
            float* pX  = out + oXs + bt * 3;
            float* pXd = out + oXd + bt * 3;
            float* pR  = out + oR  + bt * 9;
            float* pOm = out + oOm + bt * 3;
            pX[0] = x0;  pX[1] = x1;  pX[2] = x2;
            pXd[0] = xd0; pXd[1] = xd1; pXd[2] = xd2;
            pOm[0] = om0; pOm[1] = om1; pOm[2] = om2;
            #pragma unroll
            for (int i = 0; i < 3; ++i)
                #pragma unroll
                for (int j = 0; j < 3; ++j)
                    pR[i * 3 + j] = R[i][j];
        }
    }
}

extern "C" void kernel_launch(void* const* d_in, const int* in_sizes, int n_in,
                              void* d_out, int out_size, void* d_ws, size_t ws_size,
                              hipStream_t stream) {
    (void)in_sizes; (void)n_in; (void)out_size; (void)d_ws; (void)ws_size;
    const float* zg   = (const float*)d_in[0];
    const float* kg   = (const float*)d_in[1];
    const float* cg   = (const float*)d_in[2];
    const float* mug  = (const float*)d_in[3];
    const float* ctrl = (const float*)d_in[4];
    const float* xpts = (const float*)d_in[5];
    // d_in[6] (driving_mask) is recomputed on-device from x_points (dtype-agnostic)
    float* out = (float*)d_out;

    const size_t smemBytes = (4 * (size_t)kHW + 64) * sizeof(float);  // 256 KB grids + scratch
    dphysics_sim<<<dim3(kB), dim3(kP), smemBytes, stream>>>(zg, kg, cg, mug, ctrl, xpts, out);
}